// MyModel_61933428409440
// MI455X (gfx1250) — compile-verified
//
#include <hip/hip_runtime.h>

typedef __attribute__((ext_vector_type(2))) float v2f;
typedef __attribute__((ext_vector_type(8))) float v8f;

#define T_LEN 2048
#define B_SZ  512
#define I_SZ  10
#define H_SZ  20
#define G_SZ  80
#define BT    16   // batch tile (WMMA M)

__device__ __forceinline__ float fast_rcp(float x) {
#if __has_builtin(__builtin_amdgcn_rcpf)
    return __builtin_amdgcn_rcpf(x);     // v_rcp_f32 (1 ulp), avoids div expansion
#else
    return 1.0f / x;
#endif
}

__device__ __forceinline__ float fast_tanh(float x) {
#if __has_builtin(__builtin_amdgcn_tanhf)
    return __builtin_amdgcn_tanhf(x);    // native v_tanh_f32 TRANS op on gfx1250
#else
    x = fminf(15.0f, fmaxf(-15.0f, x));
    const float e = __expf(2.0f * x);
    return 1.0f - 2.0f * fast_rcp(e + 1.0f);
#endif
}

__device__ __forceinline__ float fast_sigmoid(float x) {
#if __has_builtin(__builtin_amdgcn_tanhf)
    return fmaf(0.5f, __builtin_amdgcn_tanhf(0.5f * x), 0.5f);  // 1 TRANS + mul + fma
#else
    return fast_rcp(1.0f + __expf(-x));                          // v_exp + v_rcp
#endif
}

// One wave32 per (batch-tile, direction) chain. grid = 64 blocks of 32 threads.
__global__ __launch_bounds__(32) void lstm_bidir_wmma_kernel(
    const float* __restrict__ x,
    const float* __restrict__ w_ih_f, const float* __restrict__ w_hh_f,
    const float* __restrict__ b_ih_f, const float* __restrict__ b_hh_f,
    const float* __restrict__ w_ih_r, const float* __restrict__ w_hh_r,
    const float* __restrict__ b_ih_r, const float* __restrict__ b_hh_r,
    float* __restrict__ out)
{
    __shared__ float s_hx[BT * 32];     // [m][k]: k 0..19 = h, 20..29 = x, 30..31 = zero pad
    __shared__ float s_gT[G_SZ * BT];   // [n][m]: transposed gate tiles

    const int l   = threadIdx.x;        // 0..31
    const int lm  = l & 15;
    const int hi  = l >> 4;             // lane half (A/B/D K- or M-half select)
    const int dir = blockIdx.x >> 5;    // 0 = forward, 1 = reverse
    const int bt  = blockIdx.x & 31;
    const int b0  = bt * BT;

    const float* w_ih = dir ? w_ih_r : w_ih_f;
    const float* w_hh = dir ? w_hh_r : w_hh_f;
    const float* b_ih = dir ? b_ih_r : b_ih_f;
    const float* b_hh = dir ? b_hh_r : b_hh_f;

    // ---- Fused weight W[k][n] = (k<20 ? w_hh[n][k] : k<30 ? w_ih[n][k-20] : 0)
    // held in registers as WMMA B-fragments: lane {n = tile*16+lm}, pair (k,k+1), k = 4*ks + 2*hi
    v2f   w[8][5];
    float bias[5];
    #pragma unroll
    for (int ks = 0; ks < 8; ++ks) {
        const int k = ks * 4 + 2 * hi;
        #pragma unroll
        for (int tile = 0; tile < 5; ++tile) {
            const int n = tile * 16 + lm;
            v2f wv;
            if (k < 20)      { wv[0] = w_hh[n * 20 + k];      wv[1] = w_hh[n * 20 + k + 1]; }
            else if (k < 30) { wv[0] = w_ih[n * 10 + k - 20]; wv[1] = w_ih[n * 10 + k - 19]; }
            else             { wv[0] = 0.0f;                  wv[1] = 0.0f; }
            w[ks][tile] = wv;
        }
    }
    #pragma unroll
    for (int tile = 0; tile < 5; ++tile) {
        const int n = tile * 16 + lm;
        bias[tile] = b_ih[n] + b_hh[n];
    }

    // zero h-region and pad columns (h0 = 0); x-region overwritten every step
    #pragma unroll
    for (int i = 0; i < BT; ++i) s_hx[i * 32 + l] = 0.0f;
    __syncthreads();

    float c[10];                         // cell state: lane (m=lm, hid=10*hi+j)
    #pragma unroll
    for (int j = 0; j < 10; ++j) c[j] = 0.0f;

    // x staging mapping: lane -> (row m = l/2, cols xj..xj+4)
    const int xm = l >> 1;
    const int xj = 5 * (l & 1);
    float xr[5];
    {
        const int tt0 = dir ? (T_LEN - 1) : 0;
        const float* xp = x + ((size_t)tt0 * B_SZ + b0 + xm) * I_SZ + xj;
        #pragma unroll
        for (int i = 0; i < 5; ++i) xr[i] = xp[i];
    }

    const int out_ch = dir * H_SZ;

    for (int t = 0; t < T_LEN; ++t) {
        const int tt = dir ? (T_LEN - 1 - t) : t;

        // stage this step's x slice into the K=20..29 columns
        #pragma unroll
        for (int i = 0; i < 5; ++i) s_hx[xm * 32 + 20 + xj + i] = xr[i];
        __syncthreads();

        // gates[m][n] = [h|x] @ W + bias : 8 K-steps x 5 N-tiles of V_WMMA_F32_16X16X4_F32
        v8f acc[5];
        #pragma unroll
        for (int tile = 0; tile < 5; ++tile) {
            v8f a0;
            #pragma unroll
            for (int r = 0; r < 8; ++r) a0[r] = bias[tile];
            acc[tile] = a0;
        }
        #pragma unroll
        for (int ks = 0; ks < 8; ++ks) {
            // A frag: lane {M=lm}, (K, K+1) = 4*ks + 2*hi (+1); 8-byte aligned b64 LDS load
            const v2f a = *(const v2f*)&s_hx[lm * 32 + ks * 4 + 2 * hi];
            #pragma unroll
            for (int tile = 0; tile < 5; ++tile) {
                acc[tile] = __builtin_amdgcn_wmma_f32_16x16x4_f32(
                    false, a, false, w[ks][tile], (short)0, acc[tile], false, false);
            }
        }

        // fetch next step's x into registers while WMMA drains; prefetch t+2
        {
            const int tn = dir ? max(T_LEN - 2 - t, 0) : min(t + 1, T_LEN - 1);
            const float* xp = x + ((size_t)tn * B_SZ + b0 + xm) * I_SZ + xj;
            #pragma unroll
            for (int i = 0; i < 5; ++i) xr[i] = xp[i];
            __builtin_prefetch(xp + (dir ? -(B_SZ * I_SZ) : (B_SZ * I_SZ)), 0, 1);
        }

        // scatter D tiles transposed: lane holds D[r] = gates[m=r+8*hi][n=tile*16+lm]
        // -> s_gT[n][8*hi + r], two b128 stores per tile
        #pragma unroll
        for (int tile = 0; tile < 5; ++tile) {
            const int n = tile * 16 + lm;
            float4* p = (float4*)&s_gT[n * BT + 8 * hi];
            p[0] = make_float4(acc[tile][0], acc[tile][1], acc[tile][2], acc[tile][3]);
            p[1] = make_float4(acc[tile][4], acc[tile][5], acc[tile][6], acc[tile][7]);
        }
        __syncthreads();

        // elementwise LSTM cell: lane -> (m = lm, hid = 10*hi + j)
        const int m = lm;
        float* op = out + ((size_t)tt * B_SZ + b0 + m) * (2 * H_SZ) + out_ch + 10 * hi;
        #pragma unroll
        for (int j = 0; j < 10; ++j) {
            const int hd = 10 * hi + j;
            const float ig = fast_sigmoid(s_gT[(     hd) * BT + m]);
            const float fg = fast_sigmoid(s_gT[(20 + hd) * BT + m]);
            const float gg = fast_tanh  (s_gT[(40 + hd) * BT + m]);
            const float og = fast_sigmoid(s_gT[(60 + hd) * BT + m]);
            const float cn = fmaf(fg, c[j], ig * gg);
            c[j] = cn;
            const float h = og * fast_tanh(cn);
            s_hx[m * 32 + hd] = h;       // write back for next step's A fragment
            op[j] = h;                   // stream result to HBM (no wait needed)
        }
        __syncthreads();
    }
}

extern "C" void kernel_launch(void* const* d_in, const int* in_sizes, int n_in,
                              void* d_out, int out_size, void* d_ws, size_t ws_size,
                              hipStream_t stream) {
    (void)in_sizes; (void)n_in; (void)out_size; (void)d_ws; (void)ws_size;
    const float* x      = (const float*)d_in[0];
    const float* w_ih_f = (const float*)d_in[1];
    const float* w_hh_f = (const float*)d_in[2];
    const float* b_ih_f = (const float*)d_in[3];
    const float* b_hh_f = (const float*)d_in[4];
    const float* w_ih_r = (const float*)d_in[5];
    const float* w_hh_r = (const float*)d_in[6];
    const float* b_ih_r = (const float*)d_in[7];
    const float* b_hh_r = (const float*)d_in[8];
    float* out = (float*)d_out;

    // 32 batch tiles x 2 directions = 64 independent recurrent chains, 1 wave each
    lstm_bidir_wmma_kernel<<<dim3(64), dim3(32), 0, stream>>>(
        x, w_ih_f, w_hh_f, b_ih_f, b_hh_f, w_ih_r, w_hh_r, b_ih_r, b_hh_r, out);
}